// OT_Attn_assem_18485539242602
// MI455X (gfx1250) — compile-verified
//
#include <hip/hip_runtime.h>

typedef float v2f __attribute__((ext_vector_type(2)));
typedef float v8f __attribute__((ext_vector_type(8)));

#define BS 8
#define NN 1024
#define DD 512
#define NM (NN * NN)
#define NCHUNK 8

// ---------------------------------------------------------------------------
// 1) Normalize rows of x and y (mean / std ddof=1), also emit row sum of xn^2
// ---------------------------------------------------------------------------
__global__ void normalize_kernel(const float* __restrict__ x,
                                 const float* __restrict__ y,
                                 float* __restrict__ xn, float* __restrict__ yn,
                                 float* __restrict__ x2, float* __restrict__ y2) {
  int row = blockIdx.x;                         // 0..16383 (x rows then y rows)
  const float* src;
  float* dst;
  float* s2;
  if (row < BS * NN) {
    src = x + (size_t)row * DD; dst = xn + (size_t)row * DD; s2 = x2 + row;
  } else {
    int r = row - BS * NN;
    src = y + (size_t)r * DD; dst = yn + (size_t)r * DD; s2 = y2 + r;
  }
  int t = threadIdx.x;                          // 256 threads, 2 elems each
  float a0 = src[t], a1 = src[t + 256];
  __shared__ float sh[256], sh2[256];
  sh[t] = a0 + a1;
  sh2[t] = a0 * a0 + a1 * a1;
  __syncthreads();
  for (int off = 128; off > 0; off >>= 1) {
    if (t < off) { sh[t] += sh[t + off]; sh2[t] += sh2[t + off]; }
    __syncthreads();
  }
  float mean = sh[0] * (1.0f / DD);
  float var  = (sh2[0] - sh[0] * mean) * (1.0f / (DD - 1));   // ddof=1
  float inv  = 1.0f / sqrtf(var);
  float n0 = (a0 - mean) * inv;
  float n1 = (a1 - mean) * inv;
  dst[t] = n0; dst[t + 256] = n1;
  __syncthreads();
  sh[t] = n0 * n0 + n1 * n1;
  __syncthreads();
  for (int off = 128; off > 0; off >>= 1) {
    if (t < off) sh[t] += sh[t + off];
    __syncthreads();
  }
  if (t == 0) *s2 = sh[0];
}

// ---------------------------------------------------------------------------
// 2) cost[b,n,m] = max(x2[n] + y2[m] - 2*dot(xn_n, yn_m), 0)  via f32 WMMA
//    Each wave computes one 16x16 tile; K-loop of 16x16x4 f32 WMMAs.
//    Also reduces per-batch max into cmaxbits (uint-bit atomicMax, cost>=0).
// ---------------------------------------------------------------------------
__global__ void cost_wmma_kernel(const float* __restrict__ xn,
                                 const float* __restrict__ yn,
                                 const float* __restrict__ x2,
                                 const float* __restrict__ y2,
                                 float* __restrict__ cost,
                                 unsigned* __restrict__ cmaxbits) {
  int b    = blockIdx.y;
  int lane = threadIdx.x & 31;
  int wave = threadIdx.x >> 5;
  int n0 = (blockIdx.x >> 3) << 4;                      // 64 row tiles
  int m0 = (((blockIdx.x & 7) << 3) + wave) << 4;       // 64 col tiles (8 waves/blk)

  // A (16x4 f32): lane<16 holds K={0,1}, lane>=16 holds K={2,3} for row M=lane&15.
  // B (4x16 f32): same lane pattern over yn rows (B[k][m] = yn[m][k]).
  const float* xa = xn + ((size_t)b * NN + n0 + (lane & 15)) * DD + ((lane >> 4) << 1);
  const float* yb = yn + ((size_t)b * NN + m0 + (lane & 15)) * DD + ((lane >> 4) << 1);

  v8f acc = {};
#pragma unroll 4
  for (int k = 0; k < DD; k += 4) {
    v2f av, bv;
    av.x = xa[k]; av.y = xa[k + 1];
    bv.x = yb[k]; bv.y = yb[k + 1];
    acc = __builtin_amdgcn_wmma_f32_16x16x4_f32(false, av, false, bv,
                                                (short)0, acc, false, false);
  }

  // C/D layout: VGPR r, lanes 0-15 -> M=r; lanes 16-31 -> M=8+r; N=lane&15
  int colm    = m0 + (lane & 15);
  int rowbase = n0 + ((lane >> 4) << 3);
  float y2m = y2[b * NN + colm];
  float lmax = 0.0f;
  float* cb = cost + (size_t)b * NM;
#pragma unroll
  for (int r = 0; r < 8; r++) {
    int rown = rowbase + r;
    float c = x2[b * NN + rown] + y2m - 2.0f * acc[r];
    c = fmaxf(c, 0.0f);
    cb[(size_t)rown * NN + colm] = c;
    lmax = fmaxf(lmax, c);
  }
  __shared__ float sm[256];
  sm[threadIdx.x] = lmax;
  __syncthreads();
  for (int off = 128; off > 0; off >>= 1) {
    if (threadIdx.x < off) sm[threadIdx.x] = fmaxf(sm[threadIdx.x], sm[threadIdx.x + off]);
    __syncthreads();
  }
  if (threadIdx.x == 0) atomicMax(&cmaxbits[b], __float_as_uint(sm[0]));
}

// ---------------------------------------------------------------------------
// 3) K[b,n,m] = exp(-(cost/cmax)/reg) ; f32 divide (like ref), f64 exp, f32 store
// ---------------------------------------------------------------------------
__global__ void kmat_kernel(const float* __restrict__ cost,
                            const unsigned* __restrict__ cmaxbits,
                            float* __restrict__ Km) {
  size_t idx = (size_t)blockIdx.x * 256 + threadIdx.x;
  int b = (int)(idx >> 20);
  float cmax = __uint_as_float(cmaxbits[b]);
  float Mf = cost[idx] / cmax;                    // f32, matches ref order
  Km[idx] = (float)exp(-(double)Mf * 10.0);       // /reg, reg = 0.1
}

// ---------------------------------------------------------------------------
// 4a) Kv then u = (a/Kv)^fi   (one wave32 per row, f64 accumulate)
// ---------------------------------------------------------------------------
__global__ void kv_u_kernel(const float* __restrict__ Km,
                            const double* __restrict__ v,
                            double* __restrict__ u) {
  int row  = blockIdx.x * 8 + (threadIdx.x >> 5);   // 0..8191
  int lane = threadIdx.x & 31;
  int b = row >> 10;
  const float*  krow = Km + (size_t)row * NN;
  const double* vb   = v + ((size_t)b << 10);
  double s = 0.0;
#pragma unroll
  for (int it = 0; it < 8; ++it) {
    int m = it * 128 + lane * 4;
    float4 k4 = *(const float4*)(krow + m);
    s += (double)k4.x * vb[m]     + (double)k4.y * vb[m + 1]
       + (double)k4.z * vb[m + 2] + (double)k4.w * vb[m + 3];
  }
  for (int off = 16; off > 0; off >>= 1) s += __shfl_down(s, off, 32);
  if (lane == 0) {
    const double a  = 1.0 / (double)NN;
    const double fi = 0.5 / (0.5 + 0.1);            // reg_m/(reg_m+reg)
    u[row] = pow(a / s, fi);
  }
}

// ---------------------------------------------------------------------------
// 4b) Partial K^T u over n-chunks (deterministic split, coalesced over m)
// ---------------------------------------------------------------------------
__global__ void ktu_part_kernel(const float* __restrict__ Km,
                                const double* __restrict__ u,
                                double* __restrict__ part) {
  int c = blockIdx.x * 256 + threadIdx.x;           // 0..8191 (b*1024+m)
  int chunk = blockIdx.y;                           // 0..NCHUNK-1
  int b = c >> 10;
  int m = c & 1023;
  const float*  kb = Km + ((size_t)b << 20);
  const double* ub = u + ((size_t)b << 10);
  double s = 0.0;
  int n0 = chunk * (NN / NCHUNK);
#pragma unroll 4
  for (int n = n0; n < n0 + NN / NCHUNK; n++)
    s += (double)kb[(size_t)n * NN + m] * ub[n];
  part[(size_t)chunk * (BS * NN) + c] = s;
}

// 4c) Reduce partials, v = (bmarg/Ktu)^fi
__global__ void v_update_kernel(const double* __restrict__ part,
                                double* __restrict__ v) {
  int c = blockIdx.x * 256 + threadIdx.x;
  double s = 0.0;
#pragma unroll
  for (int ch = 0; ch < NCHUNK; ch++) s += part[(size_t)ch * (BS * NN) + c];
  const double bm = 1.0 / (double)NN;
  const double fi = 0.5 / (0.5 + 0.1);
  v[c] = pow(bm / s, fi);
}

// ---------------------------------------------------------------------------
// 5) flow = u*K*v (f32 out) ; per-block f64 partial of sum(cost*flow)
// ---------------------------------------------------------------------------
__global__ void flow_dist_kernel(const float* __restrict__ Km,
                                 const float* __restrict__ cost,
                                 const double* __restrict__ u,
                                 const double* __restrict__ v,
                                 float* __restrict__ flow,
                                 double* __restrict__ distpart) {
  size_t base = ((size_t)blockIdx.x * 256 + threadIdx.x) * 4;   // 4 elems/thread
  int b = (int)(base >> 20);
  int n = (int)((base >> 10) & 1023);
  int m = (int)(base & 1023);
  float4 k4 = *(const float4*)(Km + base);
  float4 c4 = *(const float4*)(cost + base);
  double un = u[(b << 10) + n];
  const double* vb = v + ((size_t)b << 10);
  float f0 = (float)(un * (double)k4.x * vb[m]);
  float f1 = (float)(un * (double)k4.y * vb[m + 1]);
  float f2 = (float)(un * (double)k4.z * vb[m + 2]);
  float f3 = (float)(un * (double)k4.w * vb[m + 3]);
  float4 f4 = make_float4(f0, f1, f2, f3);
  *(float4*)(flow + base) = f4;
  double lsum = (double)c4.x * f0 + (double)c4.y * f1
              + (double)c4.z * f2 + (double)c4.w * f3;
  __shared__ double sh[256];
  sh[threadIdx.x] = lsum;
  __syncthreads();
  for (int off = 128; off > 0; off >>= 1) {
    if (threadIdx.x < off) sh[threadIdx.x] += sh[threadIdx.x + off];
    __syncthreads();
  }
  if (threadIdx.x == 0) distpart[blockIdx.x] = sh[0];   // 1024 blocks / batch
}

__global__ void dist_final_kernel(const double* __restrict__ distpart,
                                  float* __restrict__ dist) {
  int b = blockIdx.x;                                   // 8 blocks
  int t = threadIdx.x;
  double s = 0.0;
  for (int i = t; i < 1024; i += 256) s += distpart[b * 1024 + i];
  __shared__ double sh[256];
  sh[t] = s;
  __syncthreads();
  for (int off = 128; off > 0; off >>= 1) {
    if (t < off) sh[t] += sh[t + off];
    __syncthreads();
  }
  if (t == 0) dist[b] = (float)sh[0];
}

__global__ void init_kernel(unsigned* __restrict__ cmaxbits,
                            double* __restrict__ v) {
  int i = blockIdx.x * 256 + threadIdx.x;
  if (i < BS) cmaxbits[i] = 0u;
  if (i < BS * NN) v[i] = 1.0 / (double)NN;
}

// ---------------------------------------------------------------------------
extern "C" void kernel_launch(void* const* d_in, const int* in_sizes, int n_in,
                              void* d_out, int out_size, void* d_ws, size_t ws_size,
                              hipStream_t stream) {
  (void)in_sizes; (void)n_in; (void)out_size; (void)ws_size;
  const float* x = (const float*)d_in[0];
  const float* y = (const float*)d_in[1];

  char* ws = (char*)d_ws;
  float*    xn       = (float*)(ws);                                   // 16 MB
  float*    yn       = (float*)(ws + (16ull << 20));                   // 16 MB
  float*    x2       = (float*)(ws + (32ull << 20));                   // 32 KB
  float*    y2       = (float*)(ws + (32ull << 20) + 32768);           // 32 KB
  unsigned* cmaxbits = (unsigned*)(ws + (32ull << 20) + 65536);        // 32 B
  double*   u        = (double*)(ws + (32ull << 20) + 131072);         // 64 KB
  double*   vv       = (double*)(ws + (32ull << 20) + 196608);         // 64 KB
  double*   part     = (double*)(ws + (32ull << 20) + 262144);         // 512 KB
  double*   distpart = (double*)(ws + (32ull << 20) + 786432);         // 64 KB
  float*    cost     = (float*)(ws + (33ull << 20));                   // 32 MB
  float*    Km       = (float*)(ws + (65ull << 20));                   // 32 MB

  float* flow = (float*)d_out;
  float* dist = (float*)d_out + (size_t)BS * NM;

  init_kernel<<<32, 256, 0, stream>>>(cmaxbits, vv);
  normalize_kernel<<<2 * BS * NN, 256, 0, stream>>>(x, y, xn, yn, x2, y2);
  cost_wmma_kernel<<<dim3(512, BS), 256, 0, stream>>>(xn, yn, x2, y2, cost, cmaxbits);
  kmat_kernel<<<(BS * NM) / 256, 256, 0, stream>>>(cost, cmaxbits, Km);

  for (int it = 0; it < 500; ++it) {
    kv_u_kernel<<<(BS * NN) / 8, 256, 0, stream>>>(Km, vv, u);
    ktu_part_kernel<<<dim3((BS * NN) / 256, NCHUNK), 256, 0, stream>>>(Km, u, part);
    v_update_kernel<<<(BS * NN) / 256, 256, 0, stream>>>(part, vv);
  }

  flow_dist_kernel<<<(BS * NM) / 1024, 256, 0, stream>>>(Km, cost, u, vv, flow, distpart);
  dist_final_kernel<<<BS, 256, 0, stream>>>(distpart, dist);
}